// DifferentialNoise_64690797413085
// MI455X (gfx1250) — compile-verified
//
#include <hip/hip_runtime.h>

// DifferentialNoise: out[2t] = x[2t]; out[2t+1] = x[2t+1] - x[2t]/50.
// Memory-bound streaming kernel; math routed through V_WMMA_F32_16X16X4_F32
// as OUT_tile^T = P^T * T^T with T the constant block-bidiagonal pair matrix.

typedef __attribute__((ext_vector_type(2))) float v2f;
typedef __attribute__((ext_vector_type(8))) float v8f;

// T^T[k][n]: 1 on the diagonal; -1/50 at (k even, n = k+1); else 0.
__device__ __forceinline__ float ttc(int k, int n) {
    if (n == k) return 1.0f;
    if (((k & 1) == 0) && (n == k + 1)) return -(1.0f / 50.0f);
    return 0.0f;
}

__global__ __launch_bounds__(256) void diffnoise_wmma_kernel(
    const float* __restrict__ x, float* __restrict__ out, long long ntiles) {
    const int lane = threadIdx.x & 31;
    const int half = lane >> 4;   // 0: lanes 0-15, 1: lanes 16-31
    const int l16  = lane & 15;

    const int wavesPerBlock = blockDim.x >> 5;
    const long long gwave  = (long long)blockIdx.x * wavesPerBlock + (threadIdx.x >> 5);
    const long long nwaves = (long long)gridDim.x * wavesPerBlock;

    // Constant B operands: rows k = 4s + {0,1} (lanes 0-15) / 4s + {2,3} (lanes 16-31),
    // column n = lane % 16  (mirrors the documented 32-bit A 16x4 lane/VGPR mapping).
    v2f B0, B1, B2, B3;
    {
        const int kk = half * 2;
        B0.x = ttc(0 + kk, l16);  B0.y = ttc(1 + kk, l16);
        B1.x = ttc(4 + kk, l16);  B1.y = ttc(5 + kk, l16);
        B2.x = ttc(8 + kk, l16);  B2.y = ttc(9 + kk, l16);
        B3.x = ttc(12 + kk, l16); B3.y = ttc(13 + kk, l16);
    }

    // A operand addressing (documented 32-bit A 16x4 layout), tile = 256 floats:
    //   lane l<16 : step s loads mem[l*16 + 4s + {0,1}]
    //   lane l>=16: step s loads mem[(l-16)*16 + 4s + {2,3}]
    const int abase = l16 * 16 + half * 2;
    // D layout: VGPR v, this lane holds out element (M = v + 8*half, N = l16)
    const int dbase = (half * 8) * 16 + l16;

    for (long long t = gwave; t < ntiles; t += nwaves) {
        const float* __restrict__ src = x + t * 256;
        float* __restrict__ dst = out + t * 256;

        v2f a0 = __builtin_nontemporal_load((const v2f*)(src + abase + 0));
        v2f a1 = __builtin_nontemporal_load((const v2f*)(src + abase + 4));
        v2f a2 = __builtin_nontemporal_load((const v2f*)(src + abase + 8));
        v2f a3 = __builtin_nontemporal_load((const v2f*)(src + abase + 12));

        v8f d = {};
        // 8 args: (neg_a, A, neg_b, B, c_mod, C, reuse_a, reuse_b)
        d = __builtin_amdgcn_wmma_f32_16x16x4_f32(false, a0, false, B0, (short)0, d, false, false);
        d = __builtin_amdgcn_wmma_f32_16x16x4_f32(false, a1, false, B1, (short)0, d, false, false);
        d = __builtin_amdgcn_wmma_f32_16x16x4_f32(false, a2, false, B2, (short)0, d, false, false);
        d = __builtin_amdgcn_wmma_f32_16x16x4_f32(false, a3, false, B3, (short)0, d, false, false);

#pragma unroll
        for (int v = 0; v < 8; ++v)
            __builtin_nontemporal_store(d[v], dst + dbase + v * 16);
    }
}

// Scalar tail for sizes not a multiple of 256 (unused for this shape).
__global__ void diffnoise_tail_kernel(const float* __restrict__ x,
                                      float* __restrict__ out,
                                      long long base, long long n) {
    long long i = base + (long long)blockIdx.x * blockDim.x + threadIdx.x;
    if (i < n) {
        float v = x[i];
        out[i] = (i & 1) ? (v - x[i - 1] / 50.0f) : v;
    }
}

extern "C" void kernel_launch(void* const* d_in, const int* in_sizes, int n_in,
                              void* d_out, int out_size, void* d_ws, size_t ws_size,
                              hipStream_t stream) {
    (void)n_in; (void)d_ws; (void)ws_size; (void)in_sizes;
    const float* x = (const float*)d_in[0];
    float* out = (float*)d_out;

    const long long n = (long long)out_size;       // 2^25 for this problem
    const long long ntiles = n / 256;              // 131072 tiles of 1 KB
    const long long tail = n - ntiles * 256;

    if (ntiles > 0) {
        // 4096 blocks x 8 wave32 = 32768 waves -> 4 tiles/wave; grid-stride
        // loop is wave-uniform so EXEC stays all-ones (WMMA requirement).
        long long b = ntiles < 4096 ? ntiles : 4096;
        diffnoise_wmma_kernel<<<(int)b, 256, 0, stream>>>(x, out, ntiles);
    }
    if (tail > 0) {
        int tb = (int)((tail + 255) / 256);
        diffnoise_tail_kernel<<<tb, 256, 0, stream>>>(x, out, ntiles * 256, n);
    }
}